// WMAE_39642548142692
// MI455X (gfx1250) — compile-verified
//
#include <hip/hip_runtime.h>

typedef __attribute__((ext_vector_type(2))) float v2f;
typedef __attribute__((ext_vector_type(8))) float v8f;

// Per-feature weights from the reference module: W = {300, 1, 200}
// A float4 at element offset 4*t has weight pattern keyed by p = t % 3
// (since 4 == 1 mod 3):  p=0 -> (w0,w1,w2,w0), p=1 -> (w1,w2,w0,w1), p=2 -> (w2,w0,w1,w2)
__device__ __forceinline__ float4 weight4(int p) {
    const float4 A0 = make_float4(300.f,   1.f, 200.f, 300.f);
    const float4 A1 = make_float4(  1.f, 200.f, 300.f,   1.f);
    const float4 A2 = make_float4(200.f, 300.f,   1.f, 200.f);
    return p == 0 ? A0 : (p == 1 ? A1 : A2);
}

__device__ __forceinline__ float scalar_w(int j) {
    return j == 0 ? 300.f : (j == 1 ? 1.f : 200.f);
}

// Pass 1: grid-stride over float4s; WMMA f32 16x16x4 with A=ones acts as the
// wave accumulator (D = 1*B + C adds all 64 fp32 B-slots per instruction).
__global__ void wmae_partial_kernel(const float* __restrict__ t1,
                                    const float* __restrict__ t2,
                                    float* __restrict__ partials,
                                    int n) {
    const int n4     = n >> 2;
    const int tid    = blockIdx.x * blockDim.x + threadIdx.x;
    const int stride = gridDim.x * blockDim.x;

    const float4* __restrict__ a4 = (const float4*)t1;
    const float4* __restrict__ b4 = (const float4*)t2;

    v8f acc = {};                 // 16x16 f32 accumulator; all rows identical
    v2f ones; ones[0] = 1.f; ones[1] = 1.f;   // A matrix = all ones (16x4)
    float tail = 0.f;             // non-uniform remainder (no WMMA; EXEC may diverge)

    // Uniform trip count => EXEC is all-1s for every WMMA (ISA requirement).
    const int iters = n4 / stride;
    int i = tid;
    for (int k = 0; k < iters; ++k, i += stride) {
        float4 a = a4[i];         // global_load_b128
        float4 b = b4[i];         // global_load_b128
        float4 w = weight4(i % 3);
        float d0 = fabsf(a.x - b.x) * w.x;
        float d1 = fabsf(a.y - b.y) * w.y;
        float d2 = fabsf(a.z - b.z) * w.z;
        float d3 = fabsf(a.w - b.w) * w.w;
        v2f bv; bv[0] = d0 + d1; bv[1] = d2 + d3;
        // D = ones(16x4) x B(4x16) + C : adds every B slot into the accumulator.
        acc = __builtin_amdgcn_wmma_f32_16x16x4_f32(
            /*neg_a=*/false, ones, /*neg_b=*/false, bv,
            /*c_mod=*/(short)0, acc, /*reuse_a=*/false, /*reuse_b=*/false);
    }
    // Remainder float4s (thread-divergent): plain VALU path.
    for (; i < n4; i += stride) {
        float4 a = a4[i];
        float4 b = b4[i];
        float4 w = weight4(i % 3);
        tail += fabsf(a.x - b.x) * w.x;
        tail += fabsf(a.y - b.y) * w.y;
        tail += fabsf(a.z - b.z) * w.z;
        tail += fabsf(a.w - b.w) * w.w;
    }
    // Leftover scalars (n % 4), handled once by the first threads of block 0.
    if (blockIdx.x == 0 && (int)threadIdx.x < (n & 3)) {
        int e = (n4 << 2) + threadIdx.x;
        tail += fabsf(t1[e] - t2[e]) * scalar_w(e % 3);
    }

    // acc[0] holds two full identical rows of D across the 32 lanes
    // => full-wave sum of acc[0] is 2x the wave total.
    float v = 0.5f * acc[0] + tail;
    #pragma unroll
    for (int off = 16; off > 0; off >>= 1)
        v += __shfl_xor(v, off, 32);          // wave32 butterfly

    __shared__ float smem[32];
    const int lane = threadIdx.x & 31;
    const int wave = threadIdx.x >> 5;
    if (lane == 0) smem[wave] = v;
    __syncthreads();
    if (threadIdx.x == 0) {
        float s = 0.f;
        const int nw = blockDim.x >> 5;
        for (int w = 0; w < nw; ++w) s += smem[w];
        partials[blockIdx.x] = s;
    }
}

// Pass 2: single block reduces the per-block partials deterministically.
__global__ void wmae_final_kernel(const float* __restrict__ partials, int count,
                                  float* __restrict__ out, float inv_n) {
    float v = 0.f;
    for (int i = threadIdx.x; i < count; i += blockDim.x) v += partials[i];
    #pragma unroll
    for (int off = 16; off > 0; off >>= 1)
        v += __shfl_xor(v, off, 32);

    __shared__ float smem[32];
    const int lane = threadIdx.x & 31;
    const int wave = threadIdx.x >> 5;
    if (lane == 0) smem[wave] = v;
    __syncthreads();
    if (threadIdx.x == 0) {
        float s = 0.f;
        const int nw = blockDim.x >> 5;
        for (int w = 0; w < nw; ++w) s += smem[w];
        out[0] = s * inv_n;
    }
}

extern "C" void kernel_launch(void* const* d_in, const int* in_sizes, int n_in,
                              void* d_out, int out_size, void* d_ws, size_t ws_size,
                              hipStream_t stream) {
    const float* t1 = (const float*)d_in[0];
    const float* t2 = (const float*)d_in[1];
    float* out      = (float*)d_out;
    float* partials = (float*)d_ws;

    const int n = in_sizes[0];          // total elements = n_sample * n_feature
    const int threads = 256;            // 8 wave32 per block
    const int blocks  = 2048;           // 3,145,728 float4s / (2048*256) = 6 uniform iters

    wmae_partial_kernel<<<blocks, threads, 0, stream>>>(t1, t2, partials, n);

    // n_sample = n / 3  =>  1/n_sample = 3/n (exact in fp32 for this shape)
    const float inv_n = 3.0f / (float)n;
    wmae_final_kernel<<<1, 256, 0, stream>>>(partials, blocks, out, inv_n);
}